// NLSPNModel_32993938768572
// MI455X (gfx1250) — compile-verified
//
#include <hip/hip_runtime.h>
#include <math.h>

// ---------------------------------------------------------------------------
// NLSPN-style deformable propagation for MI455X (gfx1250, wave32, WMMA).
//   conv(8->24, 3x3)  : V_WMMA_F32_16X16X4_F32, 16px x 32ch tiles per wave
//   postprocess       : per-pixel tanh/bilinear-conf/normalize (fused in conv k.)
//   18 prop steps     : 9-tap bilinear gather, L2-resident, clamp folded in store
// All gathers use clamped-address + arithmetic-mask so the backend emits
// straight-line global_load streams instead of exec-masked branches.
// ---------------------------------------------------------------------------

namespace {

constexpr int Bn = 4, Hc = 240, Wc = 1216;
constexpr int HW  = Hc * Wc;           // 291,840
constexpr int N   = Bn * HW;           // 1,167,360  (divisible by 256 and 16)
constexpr int KD  = 72;                // 8 in-ch * 9 taps

typedef __attribute__((ext_vector_type(2))) float v2f;
typedef __attribute__((ext_vector_type(8))) float v8f;

// Bilinear sample replicating the reference _bilinear exactly.
// Loads are from clamped (always in-range) addresses and are unconditionally
// consumed by the FMA; out-of-bounds corners are handled by zeroing the
// *weight* (cndmask), so no load ever gets sunk under a divergent branch.
__device__ __forceinline__ float bilin(const float* __restrict__ img,
                                       float yy, float xx) {
  const float hm1 = (float)(Hc - 1), wm1 = (float)(Wc - 1);
  float y0 = floorf(yy), x0 = floorf(xx);
  float wy = yy - y0, wx = xx - x0;
  float acc = 0.f;
#pragma unroll
  for (int dy = 0; dy < 2; ++dy) {
#pragma unroll
    for (int dx = 0; dx < 2; ++dx) {
      float ys = y0 + (float)dy, xs = x0 + (float)dx;
      bool valid = (ys >= 0.f) && (ys <= hm1) && (xs >= 0.f) && (xs <= wm1);
      float yc = fminf(fmaxf(ys, 0.f), hm1);
      float xc = fminf(fmaxf(xs, 0.f), wm1);
      float w = (dy ? wy : 1.f - wy) * (dx ? wx : 1.f - wx);
      w = valid ? w : 0.f;                       // zero weight, not value
      float v = img[(int)yc * Wc + (int)xc];     // always-safe gather
      acc = fmaf(v, w, acc);
    }
  }
  return acc;
}

// --- prep: pad/transpose weights (24,8,3,3) -> row-major (72,32), oc>=24 -> 0
__global__ void prep_w_kernel(const float* __restrict__ w_oa,
                              float* __restrict__ wpad) {
  int i = blockIdx.x * blockDim.x + threadIdx.x;   // 0..2303
  if (i >= KD * 32) return;
  int k = i >> 5, oc = i & 31;
  float v = 0.f;
  if (oc < 24) {
    int ic = k / 9, r = k % 9, ky = r / 3, kx = r % 3;
    v = w_oa[((oc * 8 + ic) * 3 + ky) * 3 + kx];
  }
  wpad[i] = v;
}

// --- prep: keep = 1-mask, add = mask*feat_fix
__global__ void prep_mask_kernel(const float* __restrict__ ffix,
                                 float* __restrict__ keep,
                                 float* __restrict__ addv) {
  int p = blockIdx.x * blockDim.x + threadIdx.x;
  if (p >= N) return;
  float f = ffix[p];
  bool m = f > 0.f;
  keep[p] = m ? 0.f : 1.f;
  addv[p] = m ? f : 0.f;
}

// --- initial clamp: y = feat_init*keep + add
__global__ void blend_kernel(const float* __restrict__ src,
                             const float* __restrict__ keep,
                             const float* __restrict__ addv,
                             float* __restrict__ dst) {
  int p = blockIdx.x * blockDim.x + threadIdx.x;
  if (p >= N) return;
  dst[p] = fmaf(src[p], keep[p], addv[p]);
}

// --- fused conv(3x3, 8->24) via WMMA f32 16x16x4 + per-pixel postprocess.
// One wave = 16 consecutive pixels (same row) x 32 out-channels (2 N-tiles).
__global__ __launch_bounds__(256) void conv_post_kernel(
    const float* __restrict__ gsrc,    // guidance (B,8,H,W)
    const float* __restrict__ conf,    // confidence (B,H,W)
    const float* __restrict__ wpad,    // (72,32)
    const float* __restrict__ bias,    // (24)
    const float* __restrict__ scale1,  // (1)  aff_scale_const
    float* __restrict__ out_off,       // (B,18,H,W)
    float* __restrict__ out_aff) {     // (B,9,H,W)
  __shared__ float lds[8][16][24];

  const int lane = threadIdx.x & 31;
  const int wave = threadIdx.x >> 5;
  const int tile = blockIdx.x * 8 + wave;      // 72,960 tiles total
  const int p0   = tile * 16;
  const int b    = p0 / HW;
  const int rem  = p0 - b * HW;
  const int y    = rem / Wc;
  const int x0   = rem - y * Wc;               // multiple of 16, row-local

  const int pix  = lane & 15;                  // A-matrix row (pixel)
  const int koff = (lane >> 4) << 1;           // lanes>=16 hold K=k0+2,k0+3
  const float* gb = gsrc + (size_t)b * 8 * HW;

  v8f c0 = {}; v8f c1 = {};
#pragma unroll
  for (int t = 0; t < 18; ++t) {
    const int kA = 4 * t + koff;
    const int kB = kA + 1;
    v2f a;
    {
      int ic = kA / 9, r = kA % 9, ky = r / 3, kx = r - ky * 3;
      int yy = y + ky - 1, xx = x0 + pix + kx - 1;
      bool inb = (yy >= 0) && (yy < Hc) && (xx >= 0) && (xx < Wc);
      int yc = min(max(yy, 0), Hc - 1), xc = min(max(xx, 0), Wc - 1);
      a.x = (inb ? 1.f : 0.f) * gb[(ic * Hc + yc) * Wc + xc];

      ic = kB / 9; r = kB % 9; ky = r / 3; kx = r - ky * 3;
      yy = y + ky - 1; xx = x0 + pix + kx - 1;
      inb = (yy >= 0) && (yy < Hc) && (xx >= 0) && (xx < Wc);
      yc = min(max(yy, 0), Hc - 1); xc = min(max(xx, 0), Wc - 1);
      a.y = (inb ? 1.f : 0.f) * gb[(ic * Hc + yc) * Wc + xc];
    }
    v2f b0, b1;
    b0.x = wpad[kA * 32 + pix];      b0.y = wpad[kB * 32 + pix];
    b1.x = wpad[kA * 32 + 16 + pix]; b1.y = wpad[kB * 32 + 16 + pix];
    c0 = __builtin_amdgcn_wmma_f32_16x16x4_f32(false, a, false, b0,
                                               (short)0, c0, false, false);
    c1 = __builtin_amdgcn_wmma_f32_16x16x4_f32(false, a, false, b1,
                                               (short)0, c1, false, false);
  }

  // C layout: VGPR j -> rows j (lanes 0-15) / j+8 (lanes 16-31), col = lane&15
  const int mbase = (lane >> 4) ? 8 : 0;
  const int chn   = lane & 15;
#pragma unroll
  for (int j = 0; j < 8; ++j) {
    lds[wave][mbase + j][chn] = c0[j];
    if (chn < 8) lds[wave][mbase + j][16 + chn] = c1[j];
  }
  __syncthreads();

  if (lane < 16) {
    const int xp = x0 + lane;
    float oa[24];
#pragma unroll
    for (int ch = 0; ch < 24; ++ch) oa[ch] = lds[wave][lane][ch] + bias[ch];

    const float sc   = 1.f / (scale1[0] + 1e-8f);
    const float* cim = conf + (size_t)b * HW;

    // off[j] = (oa[2j], oa[2j+1]) for the 8 non-center taps (reshape quirk)
    float aff[8]; float ssum = 0.f;
#pragma unroll
    for (int j = 0; j < 8; ++j) {
      float offy = oa[2 * j], offx = oa[2 * j + 1];
      float ca = bilin(cim, (float)y + offy, (float)xp + offx);
      float a  = tanhf(oa[16 + j]) * sc * ca;
      aff[j] = a;
      ssum += fabsf(a);
    }
    ssum = fmaxf(ssum + 1e-4f, 1.f);
    float inv = 1.f / ssum, asum = 0.f;
#pragma unroll
    for (int j = 0; j < 8; ++j) { aff[j] *= inv; asum += aff[j]; }
    float aref = 1.f - asum;

    const size_t bo = (size_t)b * 18 * HW + (size_t)y * Wc + xp;
#pragma unroll
    for (int k = 0; k < 9; ++k) {
      float oy, ox;
      if (k < 4)       { oy = oa[2 * k];     ox = oa[2 * k + 1]; }
      else if (k == 4) { oy = 0.f;           ox = 0.f;           }
      else             { oy = oa[2 * k - 2]; ox = oa[2 * k - 1]; }
      out_off[bo + (size_t)(2 * k) * HW]     = oy;
      out_off[bo + (size_t)(2 * k + 1) * HW] = ox;
    }
    const size_t ba = (size_t)b * 9 * HW + (size_t)y * Wc + xp;
#pragma unroll
    for (int k = 0; k < 9; ++k) {
      float av = (k < 4) ? aff[k] : (k == 4 ? aref : aff[k - 1]);
      out_aff[ba + (size_t)k * HW] = av;
    }
  }
}

// --- one propagation step: 9-tap deformable bilinear aggregation.
// blend_out=1 folds next step's clamp into this store.
__global__ __launch_bounds__(256) void prop_kernel(
    const float* __restrict__ src,   // clamped feat (B,H,W)
    const float* __restrict__ off,   // (B,18,H,W)
    const float* __restrict__ aff,   // (B,9,H,W)
    const float* __restrict__ keep,
    const float* __restrict__ addv,
    float* __restrict__ dst,
    int blend_out) {
  int p = blockIdx.x * blockDim.x + threadIdx.x;
  if (p >= N) return;
  int b = p / HW;
  int r = p - b * HW;
  int y = r / Wc;
  int x = r - y * Wc;
  const float* img = src + (size_t)b * HW;
  const size_t ob = (size_t)b * 18 * HW + r;
  const size_t ab = (size_t)b * 9 * HW + r;
  float acc = 0.f;
#pragma unroll
  for (int k = 0; k < 9; ++k) {
    float oy = off[ob + (size_t)(2 * k) * HW];
    float ox = off[ob + (size_t)(2 * k + 1) * HW];
    float av = aff[ab + (size_t)k * HW];
    float sy = (float)y + (float)(k / 3 - 1) + oy;
    float sx = (float)x + (float)(k % 3 - 1) + ox;
    acc = fmaf(av, bilin(img, sy, sx), acc);
  }
  if (blend_out) acc = fmaf(acc, keep[p], addv[p]);
  dst[p] = acc;
}

} // namespace

extern "C" void kernel_launch(void* const* d_in, const int* in_sizes, int n_in,
                              void* d_out, int out_size, void* d_ws, size_t ws_size,
                              hipStream_t stream) {
  (void)in_sizes; (void)n_in; (void)out_size; (void)ws_size;

  const float* feat_init  = (const float*)d_in[0];
  const float* guidance   = (const float*)d_in[1];
  const float* confidence = (const float*)d_in[2];
  const float* feat_fix   = (const float*)d_in[3];
  const float* w_oa       = (const float*)d_in[4];
  const float* b_oa       = (const float*)d_in[5];
  const float* aff_scale  = (const float*)d_in[6];

  float* out      = (float*)d_out;
  float* out_feat = out;                       // (B,1,H,W)
  float* out_off  = out + (size_t)N;           // (B,18,H,W)
  float* out_aff  = out + (size_t)19 * N;      // (B,9,H,W)

  char* ws    = (char*)d_ws;
  float* wpad = (float*)ws;                    // 72*32 floats
  float* keep = (float*)(ws + 16384);
  float* addv = keep + N;
  float* buf0 = addv + N;
  float* buf1 = buf0 + N;

  const int T = 256;
  prep_w_kernel   <<<9, T, 0, stream>>>(w_oa, wpad);
  prep_mask_kernel<<<N / T, T, 0, stream>>>(feat_fix, keep, addv);
  conv_post_kernel<<<(N / 16) / 8, T, 0, stream>>>(guidance, confidence, wpad,
                                                   b_oa, aff_scale,
                                                   out_off, out_aff);
  blend_kernel    <<<N / T, T, 0, stream>>>(feat_init, keep, addv, buf0);

  float* src = buf0;
  float* alt = buf1;
  for (int t = 0; t < 18; ++t) {
    float* dst = (t == 17) ? out_feat : alt;
    prop_kernel<<<N / T, T, 0, stream>>>(src, out_off, out_aff, keep, addv,
                                         dst, (t == 17) ? 0 : 1);
    float* tmp = src; src = alt; alt = tmp;
  }
}